// EfficientAttention_25838523252895
// MI455X (gfx1250) — compile-verified
//
#include <hip/hip_runtime.h>
#include <math.h>

typedef float v2f __attribute__((ext_vector_type(2)));
typedef float v4f __attribute__((ext_vector_type(4)));
typedef float v8f __attribute__((ext_vector_type(8)));

#define B_ 2
#define H_ 16
#define S_ 2048
#define D_ 64
#define C_ 512
#define NC_ 4

// ---------------------------------------------------------------------------
// Kernel 1: Vpref[bh, i, d] = (1/512) * sum_{j < i} sum_rows V[bh, chunk j, :, d]
// (the exact contribution of every fully-masked off-diagonal chunk)
// ---------------------------------------------------------------------------
__global__ __launch_bounds__(256) void vprefix_kernel(const float* __restrict__ V,
                                                      float* __restrict__ Vpref) {
  const int bh = blockIdx.x;            // 0..31
  const int d  = threadIdx.x & 63;
  const int rg = threadIdx.x >> 6;      // 0..3
  __shared__ float part[4][64];
  const float* Vb = V + (size_t)bh * S_ * D_;
  float carry = 0.f;
  for (int i = 0; i < NC_; ++i) {
    const float* Vc = Vb + (size_t)i * C_ * D_;
    float acc = 0.f;
    for (int r = rg; r < C_; r += 4) acc += Vc[r * D_ + d];
    part[rg][d] = acc;
    __syncthreads();
    if (rg == 0) {
      Vpref[((size_t)bh * NC_ + i) * D_ + d] = carry * (1.0f / 512.0f);
      carry += part[0][d] + part[1][d] + part[2][d] + part[3][d];
    }
    __syncthreads();
  }
}

// ---------------------------------------------------------------------------
// Kernel 2: diagonal-chunk causal attention (flash-style, f32 WMMA 16x16x4)
// grid = BH * NC * 4 strip-blocks; block = 256 threads = 8 waves;
// each wave owns a 16-row q-strip.
// ---------------------------------------------------------------------------
__global__ __launch_bounds__(256) void attn_diag_kernel(
    const float* __restrict__ Q, const float* __restrict__ K,
    const float* __restrict__ V, const float* __restrict__ Vpref,
    float* __restrict__ Out) {
  const int bh    = blockIdx.x >> 4;        // 0..31
  const int rem   = blockIdx.x & 15;
  const int chunk = rem >> 2;               // 0..3
  const int sb    = rem & 3;                // 128-row strip-block in chunk
  const int wave  = threadIdx.x >> 5;
  const int lane  = threadIdx.x & 31;
  const int half  = lane >> 4;
  const int lx    = lane & 15;
  const int r0    = sb * 128 + wave * 16;   // chunk-local q row base

  const size_t base = ((size_t)bh * S_ + (size_t)chunk * C_) * D_;
  const float* Qc = Q + base;
  const float* Kc = K + base;
  const float* Vc = V + base;

  __shared__ float smem[8 * 16 * 17];       // per-wave padded S-tile scratch
  float* sw = smem + wave * (16 * 17);

  // Q A-fragments (16x4 f32 layout), pre-scaled by 1/sqrt(D)=0.125
  v2f aq[16];
  {
    const float* qp = Qc + (size_t)(r0 + lx) * D_ + half * 2;
#pragma unroll
    for (int t = 0; t < 16; ++t) {
      v2f q = *(const v2f*)(qp + 4 * t);
      aq[t] = q * 0.125f;
    }
  }

  v8f o[4] = {};                 // O^T accumulators (dims x qrows), 4 dim-tiles
  float m = -INFINITY, l = 0.f;  // softmax stats for q-row (r0 + lx)

  const int ktmax = r0 >> 4;
  for (int kt = 0; kt <= ktmax; ++kt) {
    const int k0 = kt << 4;
    if (kt < ktmax)
      __builtin_prefetch(Kc + (size_t)(k0 + 16 + lx) * D_, 0, 3);

    // ---- S(16x16) = Q * K^T via 16 x V_WMMA_F32_16X16X4_F32 ----
    v8f s = {};
    {
      const float* kp = Kc + (size_t)(k0 + lx) * D_ + half * 2;
#pragma unroll
      for (int t = 0; t < 16; ++t) {
        v2f bk = *(const v2f*)(kp + 4 * t);
        s = __builtin_amdgcn_wmma_f32_16x16x4_f32(false, aq[t], false, bk,
                                                  (short)0, s, false, false);
      }
    }

    // causal mask only on the diagonal tile
    if (kt == ktmax) {
      const int kcol  = k0 + lx;
      const int qbase = r0 + half * 8;
#pragma unroll
      for (int r = 0; r < 8; ++r)
        if (kcol > qbase + r) s[r] = -INFINITY;
    }

    // ---- C/D layout -> row-major LDS so each lane owns q-row (lx) ----
    {
      const int rowoff = half * 8;
#pragma unroll
      for (int r = 0; r < 8; ++r)
        sw[(rowoff + r) * 17 + lx] = s[r];
    }
    asm volatile("s_wait_dscnt 0" ::: "memory");

    // ---- online softmax update (lane owns row lx; halves split columns) ----
    float x[8];
    {
      const float* rowp = sw + lx * 17 + half * 2;
#pragma unroll
      for (int t = 0; t < 4; ++t) {
        x[2 * t]     = rowp[4 * t];
        x[2 * t + 1] = rowp[4 * t + 1];
      }
    }
    float hm = x[0];
#pragma unroll
    for (int j = 1; j < 8; ++j) hm = fmaxf(hm, x[j]);
    hm = fmaxf(hm, __shfl_xor(hm, 16, 32));
    const float mnew  = fmaxf(m, hm);
    const float scale = __expf(m - mnew);
    float ps[8];
    float hsum = 0.f;
#pragma unroll
    for (int j = 0; j < 8; ++j) { ps[j] = __expf(x[j] - mnew); hsum += ps[j]; }
    l = l * scale + (hsum + __shfl_xor(hsum, 16, 32));
    m = mnew;
#pragma unroll
    for (int dt = 0; dt < 4; ++dt) o[dt] *= scale;   // lane-scalar rescale

    // ---- O^T += V^T * P^T : ps[] is already the B-fragment layout ----
    {
      const float* vb = Vc + (size_t)k0 * D_;
#pragma unroll
      for (int kk = 0; kk < 4; ++kk) {
        v2f pb = {ps[2 * kk], ps[2 * kk + 1]};
        const float* vp = vb + (size_t)(4 * kk + half * 2) * D_ + lx;
#pragma unroll
        for (int dt = 0; dt < 4; ++dt) {
          v2f av = {vp[dt * 16], vp[D_ + dt * 16]};
          o[dt] = __builtin_amdgcn_wmma_f32_16x16x4_f32(false, av, false, pb,
                                                        (short)0, o[dt], false, false);
        }
      }
    }
  }

  // ---- epilogue: normalize, add prefix-mean term, 128-bit stores ----
  const float invl = 1.0f / l;
  const float* pf = Vpref + ((size_t)bh * NC_ + chunk) * D_;
  float* op = Out + ((size_t)bh * S_ + (size_t)chunk * C_ + r0 + lx) * D_;
#pragma unroll
  for (int dt = 0; dt < 4; ++dt) {
    const int dbase = dt * 16 + half * 8;
    v4f lo, hi;
#pragma unroll
    for (int j = 0; j < 4; ++j) {
      lo[j] = o[dt][j] * invl + pf[dbase + j];
      hi[j] = o[dt][4 + j] * invl + pf[dbase + 4 + j];
    }
    *(v4f*)(op + dbase) = lo;
    *(v4f*)(op + dbase + 4) = hi;
  }
}

extern "C" void kernel_launch(void* const* d_in, const int* in_sizes, int n_in,
                              void* d_out, int out_size, void* d_ws, size_t ws_size,
                              hipStream_t stream) {
  const float* Q = (const float*)d_in[0];
  const float* K = (const float*)d_in[1];
  const float* V = (const float*)d_in[2];
  float* Out   = (float*)d_out;
  float* Vpref = (float*)d_ws;  // B*H*NC*D floats = 32 KB of scratch

  vprefix_kernel<<<B_ * H_, 256, 0, stream>>>(V, Vpref);
  attn_diag_kernel<<<B_ * H_ * NC_ * 4, 256, 0, stream>>>(Q, K, V, Vpref, Out);
  (void)in_sizes; (void)n_in; (void)out_size; (void)ws_size;
}